// GraphAttention_15822659519114
// MI455X (gfx1250) — compile-verified
//
#include <hip/hip_runtime.h>

typedef __attribute__((ext_vector_type(2))) float v2f;
typedef __attribute__((ext_vector_type(8))) float v8f;

#define DD   128          // embedding dim
#define KK   64           // neighbors
#define NS   132          // padded LDS row stride (floats): 132 mod 64 banks = 4 -> conflict-free WMMA reads
#define LN_EPS 1e-5f
#define INV_ATT_SCALE (1.0f / 15.0f)

__global__ __launch_bounds__(128) void graph_attention_kernel(
    const int*   __restrict__ node_ids,   // [B*3]
    const int*   __restrict__ nbr_ids,    // [B*3, K]
    const int*   __restrict__ nbr_mask,   // [B*3, K]
    const float* __restrict__ emb,        // [V, D]
    const float* __restrict__ gamma,      // [D]
    const float* __restrict__ beta,       // [D]
    float*       __restrict__ out)        // [B*3, D]
{
    __shared__ float s_neigh[KK * NS];    // raw gathered neighbor rows (unmasked)
    __shared__ float s_node[DD];
    __shared__ float s_att[KK];           // masked, scaled attention scores
    __shared__ float s_attout[DD];
    __shared__ float s_red[8];
    __shared__ int   s_ids[KK];
    __shared__ float s_msk[KK];

    const int bt   = blockIdx.x;          // (b,t) pair index
    const int tid  = threadIdx.x;         // 0..127
    const int wv   = tid >> 5;            // wave 0..3 (wave32)
    const int ln   = tid & 31;            // lane in wave
    const int lm   = ln & 15;             // lane mod 16
    const int half = ln >> 4;             // 0: lanes 0-15, 1: lanes 16-31

    // ---- stage 0: indices, mask, node row ----
    if (tid < KK) {
        s_ids[tid] = nbr_ids [(size_t)bt * KK + tid];
        s_msk[tid] = (float)nbr_mask[(size_t)bt * KK + tid];
    }
    {
        const int nid = node_ids[bt];
        s_node[tid] = emb[(size_t)nid * DD + tid];
    }
    __syncthreads();

    // ---- stage 1: gather 64 neighbor rows -> LDS (global_load_b128 + ds_store_b128) ----
    // wave w handles rows 16w..16w+15; each lane moves one float4 (32 lanes * 4 = 128 floats/row)
    #pragma unroll 4
    for (int i = 0; i < 16; ++i) {
        const int r = wv * 16 + i;
        const float4 v = *(const float4*)(emb + (size_t)s_ids[r] * DD + ln * 4);
        float* dst = &s_neigh[r * NS + ln * 4];
        *(float4*)dst = v;
    }
    __syncthreads();

    // ---- stage 2: att = (neigh @ node) * mask / 15 via V_WMMA_F32_16X16X4_F32 ----
    // wave w -> rows 16w..16w+15 (M=16), reduce over D=128 in K=4 steps (32 WMMAs)
    {
        const int r0 = wv * 16;
        v8f acc = {};
        #pragma unroll
        for (int k = 0; k < DD; k += 4) {
            v2f a, b;
            const float* ap = &s_neigh[(r0 + lm) * NS + k + 2 * half];
            a.x = ap[0];  a.y = ap[1];                  // A: 16x4 tile of neigh
            const float* bp = &s_node[k + 2 * half];
            b.x = bp[0];  b.y = bp[1];                  // B: node chunk broadcast to all 16 cols
            acc = __builtin_amdgcn_wmma_f32_16x16x4_f32(
                false, a, false, b, (short)0, acc, false, false);
        }
        // D layout: lanes 0-15 hold rows M=0..7 in v0..v7, lanes 16-31 rows M=8..15.
        // All 16 columns identical -> lanes lm==0 publish their column.
        if (lm == 0) {
            #pragma unroll
            for (int v = 0; v < 8; ++v) {
                const int r = r0 + half * 8 + v;
                s_att[r] = acc[v] * s_msk[r] * INV_ATT_SCALE;  // mask applied here (== masking rows)
            }
        }
    }
    __syncthreads();

    // ---- stage 3: att_out = att^T @ neigh  ([1x64]@[64x128]) via WMMA ----
    // wave w -> columns 32w..32w+31 as two 16-col tiles; A rows all replicate att.
    {
        #pragma unroll
        for (int t = 0; t < 2; ++t) {
            const int col = wv * 32 + t * 16;
            v8f acc = {};
            #pragma unroll
            for (int k = 0; k < KK; k += 4) {
                v2f a, b;
                a.x = s_att[k + 2 * half];
                a.y = s_att[k + 2 * half + 1];          // A: att chunk, identical in every row
                b.x = s_neigh[(k + 2 * half)     * NS + col + lm];
                b.y = s_neigh[(k + 2 * half + 1) * NS + col + lm];  // B: 4x16 neigh tile
                acc = __builtin_amdgcn_wmma_f32_16x16x4_f32(
                    false, a, false, b, (short)0, acc, false, false);
            }
            // row 0 of D lives in acc[0] of lanes 0-15 (column N = lm)
            if (half == 0) s_attout[col + lm] = acc[0];
        }
    }
    __syncthreads();

    // ---- stage 4: LayerNorm(node + att_out) * gamma + beta ----
    {
        const float x = s_node[tid] + s_attout[tid];
        float s = x, q = x * x;
        #pragma unroll
        for (int off = 16; off >= 1; off >>= 1) {
            s += __shfl_down(s, off, 32);
            q += __shfl_down(q, off, 32);
        }
        if (ln == 0) { s_red[wv] = s; s_red[4 + wv] = q; }
        __syncthreads();
        const float S  = s_red[0] + s_red[1] + s_red[2] + s_red[3];
        const float Q  = s_red[4] + s_red[5] + s_red[6] + s_red[7];
        const float mu = S * (1.0f / (float)DD);
        const float var = Q * (1.0f / (float)DD) - mu * mu;
        const float rstd = rsqrtf(var + LN_EPS);
        out[(size_t)bt * DD + tid] = (x - mu) * rstd * gamma[tid] + beta[tid];
    }
}

extern "C" void kernel_launch(void* const* d_in, const int* in_sizes, int n_in,
                              void* d_out, int out_size, void* d_ws, size_t ws_size,
                              hipStream_t stream) {
    (void)in_sizes; (void)n_in; (void)d_ws; (void)ws_size;
    const int*   node_ids = (const int*)  d_in[0];
    const int*   nbr_ids  = (const int*)  d_in[1];
    const int*   nbr_mask = (const int*)  d_in[2];
    const float* emb      = (const float*)d_in[3];
    const float* gamma    = (const float*)d_in[4];
    const float* beta     = (const float*)d_in[5];
    float*       out      = (float*)d_out;

    const int n_bt = out_size / DD;   // B*3 = 12288
    graph_attention_kernel<<<n_bt, 128, 0, stream>>>(
        node_ids, nbr_ids, nbr_mask, emb, gamma, beta, out);
}